// DeeperGCN_55645596287039
// MI455X (gfx1250) — compile-verified
//
#include <hip/hip_runtime.h>
#include <math.h>

// Problem constants (from reference)
#define NN   100000
#define CH   64
#define HD   128
#define NE   800000
#define N64  6400000            // NN*CH
#define EPS_MSG 1e-7f
#define EPS_LN  1e-5f

typedef float v2f __attribute__((ext_vector_type(2)));
typedef float v8f __attribute__((ext_vector_type(8)));

// D = A(16x4,f32) * B(4x16,f32) + C(16x16,f32)
static __device__ __forceinline__ v8f wmma4(v2f a, v2f b, v8f c) {
  return __builtin_amdgcn_wmma_f32_16x16x4_f32(
      /*neg_a=*/false, a, /*neg_b=*/false, b,
      /*c_mod=*/(short)0, c, /*reuse_a=*/false, /*reuse_b=*/false);
}

// ---------------- utility kernels ----------------
__global__ __launch_bounds__(256) void k_zero(float4* __restrict__ p, int n4) {
  int i = blockIdx.x * 256 + threadIdx.x;
  if (i < n4) p[i] = make_float4(0.f, 0.f, 0.f, 0.f);
}

__global__ void k_zero2(float* __restrict__ p) {
  if (threadIdx.x < 2) p[threadIdx.x] = 0.f;
}

// sum + sumsq over N64 floats -> sums[0], sums[1]
__global__ __launch_bounds__(256) void k_reduce(const float4* __restrict__ h,
                                                float* __restrict__ sums) {
  __shared__ float ss[256];
  __shared__ float sq[256];
  float s = 0.f, q = 0.f;
  for (int i = blockIdx.x * 256 + threadIdx.x; i < N64 / 4; i += gridDim.x * 256) {
    float4 v = h[i];
    s += v.x + v.y + v.z + v.w;
    q += v.x * v.x + v.y * v.y + v.z * v.z + v.w * v.w;
  }
  ss[threadIdx.x] = s; sq[threadIdx.x] = q;
  __syncthreads();
  for (int off = 128; off > 0; off >>= 1) {
    if (threadIdx.x < off) {
      ss[threadIdx.x] += ss[threadIdx.x + off];
      sq[threadIdx.x] += sq[threadIdx.x + off];
    }
    __syncthreads();
  }
  if (threadIdx.x == 0) {
    atomicAdd(&sums[0], ss[0]);
    atomicAdd(&sums[1], sq[0]);
  }
}

// out = relu(graph_ln(h)) ; PyG graph LN: (h-mu)/(std+eps)*w + b over all N*C
__global__ __launch_bounds__(256) void k_gln_apply(const float4* __restrict__ h,
                                                   const float* __restrict__ sums,
                                                   const float* __restrict__ w,
                                                   const float* __restrict__ b,
                                                   float4* __restrict__ out) {
  int i = blockIdx.x * 256 + threadIdx.x;
  if (i >= N64 / 4) return;
  const float invn = 1.f / (float)N64;
  float mu = sums[0] * invn;
  float var = sums[1] * invn - mu * mu;
  float scale = 1.f / (sqrtf(fmaxf(var, 0.f)) + EPS_LN);
  int c = (i & 15) << 2;  // channel of first lane of this float4
  float4 v = h[i];
  float r0 = (v.x - mu) * scale * w[c + 0] + b[c + 0];
  float r1 = (v.y - mu) * scale * w[c + 1] + b[c + 1];
  float r2 = (v.z - mu) * scale * w[c + 2] + b[c + 2];
  float r3 = (v.w - mu) * scale * w[c + 3] + b[c + 3];
  out[i] = make_float4(fmaxf(r0, 0.f), fmaxf(r1, 0.f), fmaxf(r2, 0.f), fmaxf(r3, 0.f));
}

// ---------------- edge-phase kernels (segment softmax aggregation) ----------------
// msg = relu(hin[src]) + eps  (always > 0, so int-bit atomicMax == float max,
// and zero-init matches where(isfinite(amax), amax, 0))
__global__ __launch_bounds__(256) void k_edge_max(const int* __restrict__ src,
                                                  const int* __restrict__ dst,
                                                  const float* __restrict__ hin,
                                                  int* __restrict__ amax) {
  int idx = blockIdx.x * 256 + threadIdx.x;
  if (idx >= NE * 16) return;
  int e = idx >> 4, g = (idx & 15) << 2;
  int s = src[e], d = dst[e];
  const float4 v = *(const float4*)(hin + (size_t)s * CH + g);
  int* p = amax + (size_t)d * CH + g;
  atomicMax(p + 0, __float_as_int(fmaxf(v.x, 0.f) + EPS_MSG));
  atomicMax(p + 1, __float_as_int(fmaxf(v.y, 0.f) + EPS_MSG));
  atomicMax(p + 2, __float_as_int(fmaxf(v.z, 0.f) + EPS_MSG));
  atomicMax(p + 3, __float_as_int(fmaxf(v.w, 0.f) + EPS_MSG));
}

__global__ __launch_bounds__(256) void k_edge_sum(const int* __restrict__ src,
                                                  const int* __restrict__ dst,
                                                  const float* __restrict__ hin,
                                                  const float* __restrict__ amax,
                                                  float* __restrict__ denom) {
  int idx = blockIdx.x * 256 + threadIdx.x;
  if (idx >= NE * 16) return;
  int e = idx >> 4, g = (idx & 15) << 2;
  int s = src[e], d = dst[e];
  const float4 v = *(const float4*)(hin + (size_t)s * CH + g);
  const float4 am = *(const float4*)(amax + (size_t)d * CH + g);
  float* p = denom + (size_t)d * CH + g;
  atomicAdd(p + 0, __expf(fmaxf(v.x, 0.f) + EPS_MSG - am.x));
  atomicAdd(p + 1, __expf(fmaxf(v.y, 0.f) + EPS_MSG - am.y));
  atomicAdd(p + 2, __expf(fmaxf(v.z, 0.f) + EPS_MSG - am.z));
  atomicAdd(p + 3, __expf(fmaxf(v.w, 0.f) + EPS_MSG - am.w));
}

__global__ __launch_bounds__(256) void k_edge_agg(const int* __restrict__ src,
                                                  const int* __restrict__ dst,
                                                  const float* __restrict__ hin,
                                                  const float* __restrict__ amax,
                                                  const float* __restrict__ denom,
                                                  float* __restrict__ agg) {
  int idx = blockIdx.x * 256 + threadIdx.x;
  if (idx >= NE * 16) return;
  int e = idx >> 4, g = (idx & 15) << 2;
  int s = src[e], d = dst[e];
  const float4 v = *(const float4*)(hin + (size_t)s * CH + g);
  const float4 am = *(const float4*)(amax + (size_t)d * CH + g);
  const float4 dn = *(const float4*)(denom + (size_t)d * CH + g);
  float m0 = fmaxf(v.x, 0.f) + EPS_MSG;
  float m1 = fmaxf(v.y, 0.f) + EPS_MSG;
  float m2 = fmaxf(v.z, 0.f) + EPS_MSG;
  float m3 = fmaxf(v.w, 0.f) + EPS_MSG;
  float* p = agg + (size_t)d * CH + g;
  atomicAdd(p + 0, m0 * __expf(m0 - am.x) / dn.x);
  atomicAdd(p + 1, m1 * __expf(m1 - am.y) / dn.y);
  atomicAdd(p + 2, m2 * __expf(m2 - am.z) / dn.z);
  atomicAdd(p + 3, m3 * __expf(m3 - am.w) / dn.w);
}

// ---------------- fused GENConv MLP (WMMA f32) ----------------
// in: h = resin + agg (32-node tile); GEMM1 64->128 (+b1), row-LN, ReLU,
// GEMM2 128->64 (+b2), optional outer residual. 8 waves per block.
__global__ __launch_bounds__(256)
void k_conv_mlp(const float* __restrict__ resin, const float* __restrict__ agg,
                const float* __restrict__ W1, const float* __restrict__ b1,
                const float* __restrict__ lnw, const float* __restrict__ lnb,
                const float* __restrict__ W2, const float* __restrict__ b2,
                const float* __restrict__ outer, float* __restrict__ out) {
  __shared__ float sAct[32 * CH];
  __shared__ float sHid[32 * HD];
  __shared__ float sb1[HD], slnw[HD], slnb[HD], sb2[CH];
  const int tid = threadIdx.x;
  const int base = blockIdx.x * 32;
  if (tid < HD) { sb1[tid] = b1[tid]; slnw[tid] = lnw[tid]; slnb[tid] = lnb[tid]; }
  if (tid < CH) sb2[tid] = b2[tid];
  for (int i = tid; i < 32 * CH; i += 256) {
    size_t idx = (size_t)base * CH + i;
    sAct[i] = resin[idx] + agg[idx];
  }
  __syncthreads();

  const int wave = tid >> 5, lane = tid & 31;
  const int half = lane >> 4, l16 = lane & 15;

  // GEMM1: [32x64] @ [64x128]
  for (int task = wave; task < 16; task += 8) {
    int rt = task >> 3, ct = task & 7;
    const float* arow = &sAct[(rt * 16 + l16) * CH];
    v8f acc = {0.f, 0.f, 0.f, 0.f, 0.f, 0.f, 0.f, 0.f};
    #pragma unroll
    for (int k0 = 0; k0 < CH; k0 += 4) {
      int k = k0 + 2 * half;
      v2f a;  a.x = arow[k];                     a.y = arow[k + 1];
      v2f bb; bb.x = W1[k * HD + ct * 16 + l16]; bb.y = W1[(k + 1) * HD + ct * 16 + l16];
      acc = wmma4(a, bb, acc);
    }
    #pragma unroll
    for (int r = 0; r < 8; r++) {
      int m = rt * 16 + r + 8 * half;
      int n = ct * 16 + l16;
      sHid[m * HD + n] = acc[r] + sb1[n];
    }
  }
  __syncthreads();

  // torch LayerNorm over 128 channels per row, then ReLU
  if (tid < 32) {
    float s = 0.f, q = 0.f;
    for (int c = 0; c < HD; c++) { float v = sHid[tid * HD + c]; s += v; q += v * v; }
    float mu = s * (1.f / HD);
    float var = q * (1.f / HD) - mu * mu;
    float inv = rsqrtf(var + EPS_LN);
    for (int c = 0; c < HD; c++) {
      float v = (sHid[tid * HD + c] - mu) * inv * slnw[c] + slnb[c];
      sHid[tid * HD + c] = fmaxf(v, 0.f);
    }
  }
  __syncthreads();

  // GEMM2: [32x128] @ [128x64]
  {
    int rt = wave >> 2, ct = wave & 3;
    const float* arow = &sHid[(rt * 16 + l16) * HD];
    v8f acc = {0.f, 0.f, 0.f, 0.f, 0.f, 0.f, 0.f, 0.f};
    #pragma unroll
    for (int k0 = 0; k0 < HD; k0 += 4) {
      int k = k0 + 2 * half;
      v2f a;  a.x = arow[k];                     a.y = arow[k + 1];
      v2f bb; bb.x = W2[k * CH + ct * 16 + l16]; bb.y = W2[(k + 1) * CH + ct * 16 + l16];
      acc = wmma4(a, bb, acc);
    }
    #pragma unroll
    for (int r = 0; r < 8; r++) {
      int m = rt * 16 + r + 8 * half;
      size_t node = (size_t)(base + m);
      int n = ct * 16 + l16;
      float v = acc[r] + sb2[n];
      if (outer) v += outer[node * CH + n];
      out[node * CH + n] = v;
    }
  }
}

// ---------------- dense head: relu(xW1+b1), relu(hW2+b2), hW3+b3 ----------------
__global__ __launch_bounds__(256)
void k_head(const float* __restrict__ xc,
            const float* __restrict__ W1, const float* __restrict__ b1,
            const float* __restrict__ W2, const float* __restrict__ b2,
            const float* __restrict__ W3, const float* __restrict__ b3,
            float* __restrict__ out) {
  __shared__ float sA[32 * CH];
  __shared__ float sH[32 * CH];
  __shared__ float sb1[CH], sb2[CH], sW3[CH * 2], sb3[2];
  const int tid = threadIdx.x;
  const int base = blockIdx.x * 32;
  if (tid < CH) { sb1[tid] = b1[tid]; sb2[tid] = b2[tid]; }
  if (tid < CH * 2) sW3[tid] = W3[tid];
  if (tid < 2) sb3[tid] = b3[tid];
  for (int i = tid; i < 32 * CH; i += 256) sA[i] = xc[(size_t)base * CH + i];
  __syncthreads();

  const int wave = tid >> 5, lane = tid & 31;
  const int half = lane >> 4, l16 = lane & 15;
  const int rt = wave >> 2, ct = wave & 3;

  {  // GEMM1 [32x64]@[64x64] + ReLU
    const float* arow = &sA[(rt * 16 + l16) * CH];
    v8f acc = {0.f, 0.f, 0.f, 0.f, 0.f, 0.f, 0.f, 0.f};
    #pragma unroll
    for (int k0 = 0; k0 < CH; k0 += 4) {
      int k = k0 + 2 * half;
      v2f a;  a.x = arow[k];                     a.y = arow[k + 1];
      v2f bb; bb.x = W1[k * CH + ct * 16 + l16]; bb.y = W1[(k + 1) * CH + ct * 16 + l16];
      acc = wmma4(a, bb, acc);
    }
    #pragma unroll
    for (int r = 0; r < 8; r++) {
      int m = rt * 16 + r + 8 * half, n = ct * 16 + l16;
      sH[m * CH + n] = fmaxf(acc[r] + sb1[n], 0.f);
    }
  }
  __syncthreads();
  {  // GEMM2 [32x64]@[64x64] + ReLU (writes back to sA)
    const float* arow = &sH[(rt * 16 + l16) * CH];
    v8f acc = {0.f, 0.f, 0.f, 0.f, 0.f, 0.f, 0.f, 0.f};
    #pragma unroll
    for (int k0 = 0; k0 < CH; k0 += 4) {
      int k = k0 + 2 * half;
      v2f a;  a.x = arow[k];                     a.y = arow[k + 1];
      v2f bb; bb.x = W2[k * CH + ct * 16 + l16]; bb.y = W2[(k + 1) * CH + ct * 16 + l16];
      acc = wmma4(a, bb, acc);
    }
    #pragma unroll
    for (int r = 0; r < 8; r++) {
      int m = rt * 16 + r + 8 * half, n = ct * 16 + l16;
      sA[m * CH + n] = fmaxf(acc[r] + sb2[n], 0.f);
    }
  }
  __syncthreads();
  if (tid < 64) {  // final [32x64]@[64x2]
    int m = tid >> 1, o = tid & 1;
    float s = sb3[o];
    for (int k = 0; k < CH; k++) s += sA[m * CH + k] * sW3[k * 2 + o];
    out[(size_t)(base + m) * 2 + o] = s;
  }
}

// ---------------- launch ----------------
extern "C" void kernel_launch(void* const* d_in, const int* in_sizes, int n_in,
                              void* d_out, int out_size, void* d_ws, size_t ws_size,
                              hipStream_t stream) {
  (void)in_sizes; (void)n_in; (void)out_size; (void)ws_size;
  const float* x      = (const float*)d_in[0];
  const int*   ei     = (const int*)d_in[1];
  const int*   src    = ei;
  const int*   dst    = ei + NE;
  const float* c1_W1  = (const float*)d_in[2];
  const float* c1_b1  = (const float*)d_in[3];
  const float* c1_lnw = (const float*)d_in[4];
  const float* c1_lnb = (const float*)d_in[5];
  const float* c1_W2  = (const float*)d_in[6];
  const float* c1_b2  = (const float*)d_in[7];
  const float* n1_w   = (const float*)d_in[8];
  const float* n1_b   = (const float*)d_in[9];
  const float* cW1    = (const float*)d_in[10];
  const float* cb1    = (const float*)d_in[11];
  const float* clnw   = (const float*)d_in[12];
  const float* clnb   = (const float*)d_in[13];
  const float* cW2    = (const float*)d_in[14];
  const float* cb2    = (const float*)d_in[15];
  const float* nw     = (const float*)d_in[16];
  const float* nb     = (const float*)d_in[17];
  const float* lW1    = (const float*)d_in[18];
  const float* lb1    = (const float*)d_in[19];
  const float* lW2    = (const float*)d_in[20];
  const float* lb2    = (const float*)d_in[21];
  const float* lW3    = (const float*)d_in[22];
  const float* lb3    = (const float*)d_in[23];

  float* ws    = (float*)d_ws;
  float* xc    = ws;                        // [N,64]
  float* hpre  = ws + (size_t)N64;          // [N,64]
  float* amaxf = ws + 2 * (size_t)N64;      // [N,64] (int bits)
  int*   amax  = (int*)amaxf;
  float* denom = ws + 3 * (size_t)N64;      // [N,64]
  float* agg   = ws + 4 * (size_t)N64;      // [N,64]
  float* sums  = ws + 5 * (size_t)N64;      // [2]

  const dim3 B(256);
  const int EB   = (NE * 16 + 255) / 256;   // 50000 edge blocks
  const int Z3   = (3 * N64 / 4 + 255) / 256;
  const int APB  = (N64 / 4 + 255) / 256;
  const int MLPB = NN / 32;                 // 3125

  auto edge_phase = [&](const float* hin) {
    k_zero<<<Z3, B, 0, stream>>>((float4*)amaxf, 3 * N64 / 4);  // amax, denom, agg
    k_edge_max<<<EB, B, 0, stream>>>(src, dst, hin, amax);
    k_edge_sum<<<EB, B, 0, stream>>>(src, dst, hin, amaxf, denom);
    k_edge_agg<<<EB, B, 0, stream>>>(src, dst, hin, amaxf, denom, agg);
  };

  // Layer 1 (plain): conv on x, then xc = relu(graph_ln(h1))
  edge_phase(x);
  k_conv_mlp<<<MLPB, B, 0, stream>>>(x, agg, c1_W1, c1_b1, c1_lnw, c1_lnb,
                                     c1_W2, c1_b2, nullptr, hpre);
  k_zero2<<<1, 32, 0, stream>>>(sums);
  k_reduce<<<4096, B, 0, stream>>>((const float4*)hpre, sums);
  k_gln_apply<<<APB, B, 0, stream>>>((const float4*)hpre, sums, n1_w, n1_b, (float4*)xc);

  // Layers 2..4 (res+): h = relu(graph_ln(xc)); xc += conv(h)
  for (int i = 0; i < 3; i++) {
    k_zero2<<<1, 32, 0, stream>>>(sums);
    k_reduce<<<4096, B, 0, stream>>>((const float4*)xc, sums);
    k_gln_apply<<<APB, B, 0, stream>>>((const float4*)xc, sums, nw + i * CH, nb + i * CH,
                                       (float4*)hpre);
    edge_phase(hpre);
    k_conv_mlp<<<MLPB, B, 0, stream>>>(hpre, agg,
                                       cW1 + (size_t)i * CH * HD, cb1 + (size_t)i * HD,
                                       clnw + (size_t)i * HD, clnb + (size_t)i * HD,
                                       cW2 + (size_t)i * HD * CH, cb2 + (size_t)i * CH,
                                       xc, xc);
  }

  // Dense head
  k_head<<<MLPB, B, 0, stream>>>(xc, lW1, lb1, lW2, lb2, lW3, lb3, (float*)d_out);
}